// INF_MAMBA_87840671137990
// MI455X (gfx1250) — compile-verified
//
#include <hip/hip_runtime.h>
#include <math.h>

// ---------------------------------------------------------------------------
// Mamba U-Net forward for MI455X (gfx1250, wave32).
// GEMMs use V_WMMA_F32_16X16X4_F32 (full fp32 matrix pipe, no precision loss).
// Selective scan uses a 3-pass chunked scan (chunk=512) to break the L-length
// sequential dependency; 16-state reduction via __shfl_xor inside wave halves.
// ---------------------------------------------------------------------------

typedef float v2f __attribute__((ext_vector_type(2)));
typedef float v8f __attribute__((ext_vector_type(8)));

#define DCONV 4
#define DSTATE 16
#define CHUNK 512

__device__ __forceinline__ float siluf(float v) { return v / (1.f + expf(-v)); }
__device__ __forceinline__ float softplusf(float v) { return v > 20.f ? v : log1pf(expf(v)); }

// ---------------------------------------------------------------------------
// Generic strided GEMM:  Out[m,n] = act( sum_k X[m,k] * W[n,k] + bias[n] )
//   X indexed as X[m*ldxr + k*ldxc]   (reads channel-major maps directly)
//   Out indexed as Out[m*ldor + n*ldoc]
//   W is dense row-major (N x K) as stored in the params.
// One wave computes a 16x16 tile with V_WMMA_F32_16X16X4_F32.
//
// A (16x4): lane l<16 holds A[m=l][k0..k0+1], lane l>=16 holds A[m=l-16][k0+2..3]
// B (4x16): lane l<16 holds B[k0..k0+1][n=l], lane l>=16 holds B[k0+2..3][n=l-16]
// C/D (16x16): vgpr r of lane l = C[r + 8*(l/16)][l%16]
//
// Ragged-M / ragged-N need NO value masking: an OOB A-row feeds only an OOB
// D-row (never stored), an OOB B-col feeds only an OOB D-col (never stored).
// Addresses are merely clamped.  Only the K-tail needs zero-fill (selects).
// act: 0 none, 1 silu, 2 softplus
// ---------------------------------------------------------------------------
__global__ void gemm_wmma_f32(const float* __restrict__ X, int ldxr, int ldxc,
                              const float* __restrict__ W,
                              const float* __restrict__ bias,
                              float* __restrict__ Out, int ldor, int ldoc,
                              int M, int N, int K, int act) {
  const int lane = threadIdx.x & 31;
  const int half = lane >> 4;
  const int li   = lane & 15;
  const int n0 = blockIdx.x * 16;
  const int m0 = (blockIdx.y * blockDim.y + threadIdx.y) * 16;
  if (m0 >= M) return;  // wave-uniform exit (EXEC stays all-ones for WMMA)

  v8f acc = {};
  const int mA = m0 + li;
  const int nB = n0 + li;
  // address-clamp only; no value masking needed for M/N raggedness
  const float* __restrict__ xrow = X + (size_t)(mA < M ? mA : M - 1) * ldxr;
  const float* __restrict__ wrow = W + (size_t)(nB < N ? nB : N - 1) * K;

  int k0 = 0;
  // main loop: fully unconditional loads, 2 WMMAs per iteration
  for (; k0 + 8 <= K; k0 += 8) {
    const int ka = k0 + 2 * half;
    v2f a0, b0, a1, b1;
    a0.x = xrow[(size_t)(ka)     * ldxc];
    a0.y = xrow[(size_t)(ka + 1) * ldxc];
    b0.x = wrow[ka];
    b0.y = wrow[ka + 1];
    a1.x = xrow[(size_t)(ka + 4) * ldxc];
    a1.y = xrow[(size_t)(ka + 5) * ldxc];
    b1.x = wrow[ka + 4];
    b1.y = wrow[ka + 5];
    acc = __builtin_amdgcn_wmma_f32_16x16x4_f32(false, a0, false, b0, (short)0, acc,
                                                false, false);
    acc = __builtin_amdgcn_wmma_f32_16x16x4_f32(false, a1, false, b1, (short)0, acc,
                                                false, false);
  }
  // K tail (K%8 != 0, incl. K=2/4): clamp addresses, zero-fill overrun lanes
  for (; k0 < K; k0 += 4) {
    const int ka = k0 + 2 * half;
    const int kc0 = (ka     < K) ? ka     : K - 1;
    const int kc1 = (ka + 1 < K) ? ka + 1 : K - 1;
    float ax = xrow[(size_t)kc0 * ldxc];
    float ay = xrow[(size_t)kc1 * ldxc];
    float bx = wrow[kc0];
    float by = wrow[kc1];
    v2f a, b;
    a.x = (ka     < K) ? ax : 0.f;
    a.y = (ka + 1 < K) ? ay : 0.f;
    b.x = (ka     < K) ? bx : 0.f;
    b.y = (ka + 1 < K) ? by : 0.f;
    acc = __builtin_amdgcn_wmma_f32_16x16x4_f32(false, a, false, b, (short)0, acc,
                                                false, false);
  }

  if (nB >= N) return;
  const float bv = bias ? bias[nB] : 0.f;
#pragma unroll
  for (int r = 0; r < 8; ++r) {
    const int m = m0 + r + 8 * half;
    if (m < M) {
      float v = acc[r] + bv;
      if (act == 1) v = siluf(v);
      else if (act == 2) v = softplusf(v);
      Out[(size_t)m * ldor + (size_t)nB * ldoc] = v;
    }
  }
}

// ---------------------------------------------------------------------------
// Build 3-channel input: [img, x-coords, y-coords], channel-major (3, H*W)
// ---------------------------------------------------------------------------
__global__ void build_input_k(const float* __restrict__ img, float* __restrict__ x0,
                              int H, int W) {
  int idx = blockIdx.x * blockDim.x + threadIdx.x;
  int L = H * W;
  if (idx >= L) return;
  int w = idx % W, h = idx / W;
  x0[idx]         = img[idx];
  x0[L + idx]     = (float)w / (float)(W - 1);
  x0[2 * L + idx] = (float)h / (float)(H - 1);
}

// ---------------------------------------------------------------------------
// Generic NCHW conv2d (channel-major in/out). act: 0 none, 3 sigmoid.
// ---------------------------------------------------------------------------
__global__ void conv2d_k(const float* __restrict__ in, const float* __restrict__ w,
                         const float* __restrict__ b, float* __restrict__ out,
                         int Ci, int Hi, int Wi, int Co, int Ho, int Wo,
                         int KH, int KW, int stride, int pad, int act) {
  int idx = blockIdx.x * blockDim.x + threadIdx.x;
  int total = Co * Ho * Wo;
  if (idx >= total) return;
  int wo = idx % Wo;
  int t = idx / Wo;
  int ho = t % Ho;
  int co = t / Ho;
  float acc = b[co];
  for (int ci = 0; ci < Ci; ++ci) {
    for (int ky = 0; ky < KH; ++ky) {
      int hi = ho * stride - pad + ky;
      if (hi < 0 || hi >= Hi) continue;
      for (int kx = 0; kx < KW; ++kx) {
        int wi = wo * stride - pad + kx;
        if (wi < 0 || wi >= Wi) continue;
        acc += in[((size_t)ci * Hi + hi) * Wi + wi] *
               w[(((size_t)co * Ci + ci) * KH + ky) * KW + kx];
      }
    }
  }
  if (act == 3) acc = 1.f / (1.f + expf(-acc));
  out[idx] = acc;
}

// ---------------------------------------------------------------------------
// Transposed conv 2x2 stride 2: out[o,2h+ky,2w+kx] = sum_c x[c,h,w]*W[c,o,ky,kx]+b
// ---------------------------------------------------------------------------
__global__ void upconv2x2_k(const float* __restrict__ in, const float* __restrict__ w,
                            const float* __restrict__ b, float* __restrict__ out,
                            int Ci, int H, int W, int Co) {
  int Ho = 2 * H, Wo = 2 * W;
  int idx = blockIdx.x * blockDim.x + threadIdx.x;
  int total = Co * Ho * Wo;
  if (idx >= total) return;
  int wo = idx % Wo;
  int t = idx / Wo;
  int ho = t % Ho;
  int o = t / Ho;
  int h = ho >> 1, ky = ho & 1, ww = wo >> 1, kx = wo & 1;
  float acc = b[o];
  for (int c = 0; c < Ci; ++c)
    acc += in[((size_t)c * H + h) * W + ww] * w[(((size_t)c * Co + o) * 2 + ky) * 2 + kx];
  out[idx] = acc;
}

// ---------------------------------------------------------------------------
// Channel LayerNorm (per spatial position over C) + SiLU.  In/out (C, L).
// ---------------------------------------------------------------------------
__global__ void ln_silu_k(const float* __restrict__ xin, const float* __restrict__ g,
                          const float* __restrict__ b, float* __restrict__ xout,
                          int C, int L) {
  int l = blockIdx.x * blockDim.x + threadIdx.x;
  if (l >= L) return;
  float m = 0.f, m2 = 0.f;
  for (int c = 0; c < C; ++c) {
    float v = xin[(size_t)c * L + l];
    m += v;
    m2 += v * v;
  }
  m /= (float)C;
  float var = m2 / (float)C - m * m;
  float inv = rsqrtf(var + 1e-5f);
  for (int c = 0; c < C; ++c) {
    float v = (xin[(size_t)c * L + l] - m) * inv * g[c] + b[c];
    xout[(size_t)c * L + l] = siluf(v);
  }
}

// ---------------------------------------------------------------------------
// Depthwise causal conv1d (width 4) over sequence + SiLU.
// xz is (L, 2*di) row-major; xi = first di columns.  Output xi2 (L, di).
// ---------------------------------------------------------------------------
__global__ void dwconv_silu_k(const float* __restrict__ xz, const float* __restrict__ cw,
                              const float* __restrict__ cb, float* __restrict__ xi2,
                              int L, int di) {
  int idx = blockIdx.x * blockDim.x + threadIdx.x;
  if (idx >= L * di) return;
  int d = idx % di;
  int l = idx / di;
  float acc = cb[d];
#pragma unroll
  for (int j = 0; j < DCONV; ++j) {
    int ls = l - (DCONV - 1) + j;
    if (ls >= 0) acc += xz[(size_t)ls * (2 * di) + d] * cw[d * DCONV + j];
  }
  xi2[idx] = siluf(acc);
}

// ---------------------------------------------------------------------------
// Chunked selective scan.  Wave32 layout: lane = s + 16*(d&1), block.x -> d/2.
// h[l] = exp(delta*A)*h[l-1] + (delta*xi)*B[l]   (per d, s)
// ---------------------------------------------------------------------------
__global__ void scan_pass1(const float* __restrict__ dbc, const float* __restrict__ delta,
                           const float* __restrict__ xi2, const float* __restrict__ A_log,
                           float* __restrict__ chA, float* __restrict__ chH,
                           int L, int di, int dr) {
  int lane = threadIdx.x;
  int s = lane & 15;
  int d = blockIdx.x * 2 + (lane >> 4);
  int c = blockIdx.y;
  float A = -expf(A_log[d * DSTATE + s]);
  int nbc = dr + 2 * DSTATE;
  float aprod = 1.f, h = 0.f;
  int l0 = c * CHUNK;
  int l1 = min(L, l0 + CHUNK);
  for (int l = l0; l < l1; ++l) {
    float dl = delta[(size_t)l * di + d];
    float xv = xi2[(size_t)l * di + d];
    float bm = dbc[(size_t)l * nbc + dr + s];
    float dA = expf(dl * A);
    h = dA * h + (dl * xv) * bm;
    aprod *= dA;
  }
  size_t o = ((size_t)c * di + d) * DSTATE + s;
  chA[o] = aprod;
  chH[o] = h;
}

__global__ void scan_pass2(const float* __restrict__ chA, const float* __restrict__ chH,
                           float* __restrict__ chHin, int di, int nchunks) {
  int lane = threadIdx.x;
  int s = lane & 15;
  int d = blockIdx.x * 2 + (lane >> 4);
  float h = 0.f;
  for (int c = 0; c < nchunks; ++c) {
    size_t o = ((size_t)c * di + d) * DSTATE + s;
    chHin[o] = h;
    h = chA[o] * h + chH[o];
  }
}

__global__ void scan_pass3(const float* __restrict__ xz, const float* __restrict__ dbc,
                           const float* __restrict__ delta, const float* __restrict__ xi2,
                           const float* __restrict__ A_log, const float* __restrict__ Dp,
                           const float* __restrict__ chHin, float* __restrict__ y,
                           int L, int di, int dr) {
  int lane = threadIdx.x;
  int s = lane & 15;
  int d = blockIdx.x * 2 + (lane >> 4);
  int c = blockIdx.y;
  float A = -expf(A_log[d * DSTATE + s]);
  float Dd = Dp[d];
  int nbc = dr + 2 * DSTATE;
  float h = chHin[((size_t)c * di + d) * DSTATE + s];
  int l0 = c * CHUNK;
  int l1 = min(L, l0 + CHUNK);
  for (int l = l0; l < l1; ++l) {
    float dl = delta[(size_t)l * di + d];
    float xv = xi2[(size_t)l * di + d];
    float bm = dbc[(size_t)l * nbc + dr + s];
    float cm = dbc[(size_t)l * nbc + dr + DSTATE + s];
    h = expf(dl * A) * h + (dl * xv) * bm;
    float p = h * cm;
    // reduce over 16 states (xor masks 1..8 stay within each 16-lane half)
    p += __shfl_xor(p, 8, 32);
    p += __shfl_xor(p, 4, 32);
    p += __shfl_xor(p, 2, 32);
    p += __shfl_xor(p, 1, 32);
    if (s == 0) {
      float zz = xz[(size_t)l * (2 * di) + di + d];
      y[(size_t)l * di + d] = (p + xv * Dd) * siluf(zz);
    }
  }
}

__global__ void copy_k(const float* __restrict__ src, float* __restrict__ dst, int n) {
  int i = blockIdx.x * blockDim.x + threadIdx.x;
  if (i < n) dst[i] = src[i];
}

// ---------------------------------------------------------------------------
// Host-side orchestration
// ---------------------------------------------------------------------------
struct MambaP {
  const float *A_log, *D, *conv_b, *conv_w, *dt_b, *dt_w, *in_b, *in_w, *out_b,
      *out_w, *xproj_w;
};

static MambaP mp(void* const* d, int base) {
  MambaP p;
  p.A_log   = (const float*)d[base + 0];
  p.D       = (const float*)d[base + 1];
  p.conv_b  = (const float*)d[base + 2];
  p.conv_w  = (const float*)d[base + 3];
  p.dt_b    = (const float*)d[base + 4];
  p.dt_w    = (const float*)d[base + 5];
  p.in_b    = (const float*)d[base + 6];
  p.in_w    = (const float*)d[base + 7];
  p.out_b   = (const float*)d[base + 8];
  p.out_w   = (const float*)d[base + 9];
  p.xproj_w = (const float*)d[base + 10];
  return p;
}

static void launch_gemm(hipStream_t st, const float* X, int lxr, int lxc,
                        const float* W, const float* bias, float* O, int lor, int loc,
                        int M, int N, int K, int act) {
  dim3 blk(32, 4);
  dim3 grd((N + 15) / 16, (((M + 15) / 16) + 3) / 4);
  gemm_wmma_f32<<<grd, blk, 0, st>>>(X, lxr, lxc, W, bias, O, lor, loc, M, N, K, act);
}

// feat: input features, channel-major (C, L).  out_cl: output (C, L).
static void run_mamba(hipStream_t st, const float* feat, float* out_cl,
                      const MambaP& p, int C, int L, float* arena) {
  const int di = 2 * C;
  const int dr = (C + 15) / 16;
  const int nbc = dr + 2 * DSTATE;
  const int nch = (L + CHUNK - 1) / CHUNK;

  float* xz    = arena;
  float* xi2   = xz + (size_t)L * 2 * di;
  float* dbc   = xi2 + (size_t)L * di;
  float* delta = dbc + (size_t)L * nbc;
  float* y     = delta + (size_t)L * di;
  float* chA   = y + (size_t)L * di;
  float* chH   = chA + (size_t)nch * di * DSTATE;
  float* chHin = chH + (size_t)nch * di * DSTATE;

  // 1) in-proj: xz(L, 2di) = feat^T @ in_w^T + in_b
  launch_gemm(st, feat, 1, L, p.in_w, p.in_b, xz, 2 * di, 1, L, 2 * di, C, 0);
  // 2) depthwise causal conv1d + SiLU -> xi2 (L, di)
  {
    int n = L * di;
    dwconv_silu_k<<<(n + 255) / 256, 256, 0, st>>>(xz, p.conv_w, p.conv_b, xi2, L, di);
  }
  // 3) x-proj: dbc(L, dr+32) = xi2 @ xproj_w^T  (no bias)
  launch_gemm(st, xi2, di, 1, p.xproj_w, nullptr, dbc, nbc, 1, L, nbc, di, 0);
  // 4) dt-proj + softplus: delta(L, di) = softplus(dbc[:, :dr] @ dt_w^T + dt_b)
  launch_gemm(st, dbc, nbc, 1, p.dt_w, p.dt_b, delta, di, 1, L, di, dr, 2);
  // 5) chunked selective scan + gating -> y (L, di)
  scan_pass1<<<dim3(di / 2, nch), 32, 0, st>>>(dbc, delta, xi2, p.A_log, chA, chH, L, di, dr);
  scan_pass2<<<dim3(di / 2), 32, 0, st>>>(chA, chH, chHin, di, nch);
  scan_pass3<<<dim3(di / 2, nch), 32, 0, st>>>(xz, dbc, delta, xi2, p.A_log, p.D, chHin,
                                               y, L, di, dr);
  // 6) out-proj -> channel-major (C, L)
  launch_gemm(st, y, di, 1, p.out_w, p.out_b, out_cl, 1, L, L, C, di, 0);
}

extern "C" void kernel_launch(void* const* d_in, const int* in_sizes, int n_in,
                              void* d_out, int out_size, void* d_ws, size_t ws_size,
                              hipStream_t stream) {
  (void)in_sizes; (void)n_in; (void)out_size; (void)ws_size;
  const int H = 192, Wd = 192;
  const int L1 = H * Wd, L2 = L1 / 4, L3 = L1 / 16;

  // ---- param index map (jax tree-flatten: dict keys sorted at each level) ----
  // 0: img_v
  // 1-11: bneck mamba {A_log,D,conv_b,conv_w,dt_b,dt_w,in_b,in_w,out_b,out_w,xproj_w}
  // 12,13: bneck_norm {b,g}
  // 14-24: dec1.mamba; 25,26: dec1.norm{b,g}; 27,28: dec1.out{b,w}; 29,30: dec1.up{b,w}
  // 31-41: dec2.mamba; 42,43: dec2.norm; 44,45: dec2.out; 46,47: dec2.up
  // 48,49: enc1.down{b,w}; 50-60: enc1.mamba; 61,62: enc1.norm{b,g}
  // 63,64: enc2.down{b,w}; 65-75: enc2.mamba; 76,77: enc2.norm{b,g}
  // 78,79: input_conv{b,w}; 80,81: input_norm{b,g}; 82,83: out_conv{b,w}
  const float* img = (const float*)d_in[0];
  MambaP bkm = mp(d_in, 1);
  const float* bn_b = (const float*)d_in[12];
  const float* bn_g = (const float*)d_in[13];
  MambaP d1m = mp(d_in, 14);
  const float* d1n_b = (const float*)d_in[25];
  const float* d1n_g = (const float*)d_in[26];
  const float* d1o_b = (const float*)d_in[27];
  const float* d1o_w = (const float*)d_in[28];
  const float* d1u_b = (const float*)d_in[29];
  const float* d1u_w = (const float*)d_in[30];
  MambaP d2m = mp(d_in, 31);
  const float* d2n_b = (const float*)d_in[42];
  const float* d2n_g = (const float*)d_in[43];
  const float* d2o_b = (const float*)d_in[44];
  const float* d2o_w = (const float*)d_in[45];
  const float* d2u_b = (const float*)d_in[46];
  const float* d2u_w = (const float*)d_in[47];
  const float* e1d_b = (const float*)d_in[48];
  const float* e1d_w = (const float*)d_in[49];
  MambaP e1m = mp(d_in, 50);
  const float* e1n_b = (const float*)d_in[61];
  const float* e1n_g = (const float*)d_in[62];
  const float* e2d_b = (const float*)d_in[63];
  const float* e2d_w = (const float*)d_in[64];
  MambaP e2m = mp(d_in, 65);
  const float* e2n_b = (const float*)d_in[76];
  const float* e2n_g = (const float*)d_in[77];
  const float* ic_b = (const float*)d_in[78];
  const float* ic_w = (const float*)d_in[79];
  const float* inn_b = (const float*)d_in[80];
  const float* inn_g = (const float*)d_in[81];
  const float* oc_b = (const float*)d_in[82];
  const float* oc_w = (const float*)d_in[83];

  // ---- workspace layout (floats); persistent buffers then reusable arena ----
  float* ws = (float*)d_ws;
  float* x0   = ws;                         // (3, L1)
  float* feat = x0 + (size_t)3 * L1;        // (32, L1)
  float* r1   = feat + (size_t)32 * L1;     // skip1 (32, L1)
  float* x1   = r1 + (size_t)32 * L1;       // (64, L2)
  float* r2   = x1 + (size_t)64 * L2;       // skip2 (64, L2)
  float* x2   = r2 + (size_t)64 * L2;       // (128, L3)
  float* xb   = x2 + (size_t)128 * L3;      // bneck post-norm (128, L3)
  float* cat1 = xb + (size_t)128 * L3;      // (128, L2)
  float* xm   = cat1 + (size_t)128 * L2;    // post-ln, sized 64*L1 (covers 128*L2)
  float* xd1  = xm + (size_t)64 * L1;       // (64, L2)
  float* cat2 = xd1 + (size_t)64 * L2;      // (64, L1)
  float* xd2  = cat2 + (size_t)64 * L1;     // (32, L1)
  float* mtmp = xd2 + (size_t)32 * L1;      // raw mamba out, sized 64*L1
  float* arena = mtmp + (size_t)64 * L1;    // mamba internals (reused each block)

  const int TB = 256;
  // ---- stem: coords + 3x3 conv + ln + silu ----
  build_input_k<<<(L1 + TB - 1) / TB, TB, 0, stream>>>(img, x0, H, Wd);
  conv2d_k<<<((32 * L1) + TB - 1) / TB, TB, 0, stream>>>(x0, ic_w, ic_b, mtmp,
                                                         3, H, Wd, 32, H, Wd, 3, 3, 1, 1, 0);
  ln_silu_k<<<(L1 + TB - 1) / TB, TB, 0, stream>>>(mtmp, inn_g, inn_b, feat, 32, L1);

  // ---- enc1 (C=32 @ 192x192) ----
  run_mamba(stream, feat, mtmp, e1m, 32, L1, arena);
  ln_silu_k<<<(L1 + TB - 1) / TB, TB, 0, stream>>>(mtmp, e1n_g, e1n_b, r1, 32, L1);
  conv2d_k<<<((64 * L2) + TB - 1) / TB, TB, 0, stream>>>(r1, e1d_w, e1d_b, x1,
                                                         32, 192, 192, 64, 96, 96, 3, 3, 2, 1, 0);

  // ---- enc2 (C=64 @ 96x96) ----
  run_mamba(stream, x1, mtmp, e2m, 64, L2, arena);
  ln_silu_k<<<(L2 + TB - 1) / TB, TB, 0, stream>>>(mtmp, e2n_g, e2n_b, r2, 64, L2);
  conv2d_k<<<((128 * L3) + TB - 1) / TB, TB, 0, stream>>>(r2, e2d_w, e2d_b, x2,
                                                          64, 96, 96, 128, 48, 48, 3, 3, 2, 1, 0);

  // ---- bottleneck (C=128 @ 48x48) ----
  run_mamba(stream, x2, mtmp, bkm, 128, L3, arena);
  ln_silu_k<<<(L3 + TB - 1) / TB, TB, 0, stream>>>(mtmp, bn_g, bn_b, xb, 128, L3);

  // ---- dec1: up(128->64 @96), concat skip2, mamba(C=128 @96), ln, 1x1 ----
  upconv2x2_k<<<((64 * L2) + TB - 1) / TB, TB, 0, stream>>>(xb, d1u_w, d1u_b, cat1,
                                                            128, 48, 48, 64);
  copy_k<<<((64 * L2) + TB - 1) / TB, TB, 0, stream>>>(r2, cat1 + (size_t)64 * L2, 64 * L2);
  run_mamba(stream, cat1, mtmp, d1m, 128, L2, arena);
  ln_silu_k<<<(L2 + TB - 1) / TB, TB, 0, stream>>>(mtmp, d1n_g, d1n_b, xm, 128, L2);
  launch_gemm(stream, xm, 1, L2, d1o_w, d1o_b, xd1, 1, L2, L2, 64, 128, 0);  // 1x1 conv

  // ---- dec2: up(64->32 @192), concat skip1, mamba(C=64 @192), ln, 1x1 ----
  upconv2x2_k<<<((32 * L1) + TB - 1) / TB, TB, 0, stream>>>(xd1, d2u_w, d2u_b, cat2,
                                                            64, 96, 96, 32);
  copy_k<<<((32 * L1) + TB - 1) / TB, TB, 0, stream>>>(r1, cat2 + (size_t)32 * L1, 32 * L1);
  run_mamba(stream, cat2, mtmp, d2m, 64, L1, arena);
  ln_silu_k<<<(L1 + TB - 1) / TB, TB, 0, stream>>>(mtmp, d2n_g, d2n_b, xm, 64, L1);
  launch_gemm(stream, xm, 1, L1, d2o_w, d2o_b, xd2, 1, L1, L1, 32, 64, 0);   // 1x1 conv

  // ---- head: 3x3 conv (32->1) + sigmoid -> d_out ----
  conv2d_k<<<(L1 + TB - 1) / TB, TB, 0, stream>>>(xd2, oc_w, oc_b, (float*)d_out,
                                                  32, H, Wd, 1, H, Wd, 3, 3, 1, 1, 3);
}